// SpatialGraphEncoder_12257836663335
// MI455X (gfx1250) — compile-verified
//
#include <hip/hip_runtime.h>
#include <hip/hip_bf16.h>
#include <math.h>

#define N_FRAMES 8
#define V        10000
#define E        160000
#define C_IN     64
#define H        128
#define SLOPE    0.2f
#define EDGES_TOT (E + V)

typedef __attribute__((ext_vector_type(2))) float v2f;
typedef __attribute__((ext_vector_type(8))) float v8f;

// ---------------------------------------------------------------------------
// Dual GEMM via V_WMMA_F32_16X16X4_F32:  outl = X @ Wl,  outr = X @ Wr
// X: [V, C] row-major,  Wl/Wr: [C, H] row-major,  outl/outr: [V, H]
// grid.x = V/16 row tiles; block = 256 threads = 8 waves; wave w -> col tile w.
// V = 625*16 exactly -> no tail, EXEC all-1s for WMMA.
// ---------------------------------------------------------------------------
template<int C>
__global__ __launch_bounds__(256) void gemm_dual_wmma(
    const float* __restrict__ X, const float* __restrict__ Wl,
    const float* __restrict__ Wr, float* __restrict__ outl,
    float* __restrict__ outr) {
  const int lane  = threadIdx.x & 31;
  const int wave  = threadIdx.x >> 5;      // 0..7 -> column tile
  const int m0    = blockIdx.x * 16;       // row-tile base
  const int n0    = wave * 16;             // col-tile base
  const int row   = m0 + (lane & 15);      // A-fragment row (M)
  const int col   = n0 + (lane & 15);      // B-fragment col (N)
  const int khalf = (lane >> 4) * 2;       // lanes 0-15: K=0,1 ; lanes 16-31: K=2,3

  v8f cl = {};
  v8f cr = {};
  for (int k = 0; k < C; k += 4) {
    v2f a, bl, br;
    a[0]  = X[(size_t)row * C + k + khalf];
    a[1]  = X[(size_t)row * C + k + khalf + 1];
    bl[0] = Wl[(size_t)(k + khalf)     * H + col];
    bl[1] = Wl[(size_t)(k + khalf + 1) * H + col];
    br[0] = Wr[(size_t)(k + khalf)     * H + col];
    br[1] = Wr[(size_t)(k + khalf + 1) * H + col];
    cl = __builtin_amdgcn_wmma_f32_16x16x4_f32(false, a, false, bl, (short)0, cl, false, false);
    cr = __builtin_amdgcn_wmma_f32_16x16x4_f32(false, a, false, br, (short)0, cr, false, false);
  }

  // C/D layout: VGPR v -> M = v (+8 for lanes 16-31), N = lane & 15
  const int rbase = m0 + ((lane >> 4) << 3);
  const int cidx  = n0 + (lane & 15);
#pragma unroll
  for (int v = 0; v < 8; ++v) {
    outl[(size_t)(rbase + v) * H + cidx] = cl[v];
    outr[(size_t)(rbase + v) * H + cidx] = cr[v];
  }
}

// ---------------------------------------------------------------------------
// Init per-layer buffers: mmax = -inf, denom = 0, acc = 0
// ---------------------------------------------------------------------------
__global__ void init_buffers(float* __restrict__ mmax, float* __restrict__ denom,
                             float* __restrict__ acc) {
  int idx = blockIdx.x * blockDim.x + threadIdx.x;
  if (idx < V) {
    mmax[idx]  = __int_as_float(0xff800000);  // -inf
    denom[idx] = 0.0f;
  }
  if (idx < V * H) acc[idx] = 0.0f;
}

// ---------------------------------------------------------------------------
// Per-edge attention logit: e = att . leaky_relu(xl[src] + xr[dst])
// One wave32 per edge; shuffle reduction; float atomic-max on mmax[dst].
// Edges E..E+V-1 are self-loops (src = dst = edge - E).
// ---------------------------------------------------------------------------
__global__ __launch_bounds__(256) void edge_logit(
    const float* __restrict__ xl, const float* __restrict__ xr,
    const int* __restrict__ ei, const float* __restrict__ att,
    float* __restrict__ e_out, float* __restrict__ mmax) {
  const int lane = threadIdx.x & 31;
  const int edge = (blockIdx.x * blockDim.x + threadIdx.x) >> 5;
  if (edge >= EDGES_TOT) return;

  int src, dst;
  if (edge < E) { src = ei[edge]; dst = ei[E + edge]; }
  else          { src = dst = edge - E; }

  float acc = 0.0f;
  for (int j = lane; j < H; j += 32) {
    float v = xl[(size_t)src * H + j] + xr[(size_t)dst * H + j];
    v = (v > 0.0f) ? v : SLOPE * v;
    acc += v * att[j];
  }
  for (int off = 16; off > 0; off >>= 1)
    acc += __shfl_down(acc, off, 32);

  if (lane == 0) {
    e_out[edge] = acc;
    // IEEE-correct float atomic max with -inf init:
    // positive -> signed int max; negative -> unsigned int min.
    int* ai = (int*)&mmax[dst];
    if (acc >= 0.0f) atomicMax(ai, __float_as_int(acc));
    else             atomicMin((unsigned int*)ai, (unsigned int)__float_as_int(acc));
  }
}

// ---------------------------------------------------------------------------
// a = exp(e - m[dst]); denom[dst] += a   (thread per edge)
// ---------------------------------------------------------------------------
__global__ void edge_exp(const float* __restrict__ e_in,
                         const float* __restrict__ mmax,
                         const int* __restrict__ ei,
                         float* __restrict__ a_out, float* __restrict__ denom) {
  int edge = blockIdx.x * blockDim.x + threadIdx.x;
  if (edge >= EDGES_TOT) return;
  int dst = (edge < E) ? ei[E + edge] : edge - E;
  float a = expf(e_in[edge] - mmax[dst]);
  a_out[edge] = a;
  atomicAdd(&denom[dst], a);
}

// ---------------------------------------------------------------------------
// acc[dst] += (a/denom[dst]) * xl[src]   (wave32 per edge, f32 atomics to L2)
// ---------------------------------------------------------------------------
__global__ __launch_bounds__(256) void edge_scatter(
    const float* __restrict__ xl, const int* __restrict__ ei,
    const float* __restrict__ a_in, const float* __restrict__ denom,
    float* __restrict__ acc) {
  const int lane = threadIdx.x & 31;
  const int edge = (blockIdx.x * blockDim.x + threadIdx.x) >> 5;
  if (edge >= EDGES_TOT) return;

  int src, dst;
  if (edge < E) { src = ei[edge]; dst = ei[E + edge]; }
  else          { src = dst = edge - E; }

  float alpha = a_in[edge] / (denom[dst] + 1e-16f);
  for (int j = lane; j < H; j += 32)
    atomicAdd(&acc[(size_t)dst * H + j], alpha * xl[(size_t)src * H + j]);
}

// ---------------------------------------------------------------------------
// out = relu(acc + b)
// ---------------------------------------------------------------------------
__global__ void bias_relu(const float* __restrict__ acc,
                          const float* __restrict__ b, float* __restrict__ out) {
  int idx = blockIdx.x * blockDim.x + threadIdx.x;
  if (idx >= V * H) return;
  float v = acc[idx] + b[idx & (H - 1)];
  out[idx] = v > 0.0f ? v : 0.0f;
}

// ---------------------------------------------------------------------------
extern "C" void kernel_launch(void* const* d_in, const int* in_sizes, int n_in,
                              void* d_out, int out_size, void* d_ws, size_t ws_size,
                              hipStream_t stream) {
  const float* x    = (const float*)d_in[0];
  const int*   ei   = (const int*)  d_in[1];
  const float* Wl1  = (const float*)d_in[2];
  const float* Wr1  = (const float*)d_in[3];
  const float* att1 = (const float*)d_in[4];
  const float* b1   = (const float*)d_in[5];
  const float* Wl2  = (const float*)d_in[6];
  const float* Wr2  = (const float*)d_in[7];
  const float* att2 = (const float*)d_in[8];
  const float* b2   = (const float*)d_in[9];
  float* out = (float*)d_out;

  // Workspace partition (floats)
  float* ws    = (float*)d_ws;
  float* xl    = ws;                        // V*H
  float* xr    = xl    + (size_t)V * H;     // V*H
  float* acc   = xr    + (size_t)V * H;     // V*H
  float* h1    = acc   + (size_t)V * H;     // V*H
  float* evals = h1    + (size_t)V * H;     // EDGES_TOT
  float* avals = evals + EDGES_TOT;         // EDGES_TOT
  float* mmax  = avals + EDGES_TOT;         // V
  float* denom = mmax  + V;                 // V

  const dim3 blk(256);
  const dim3 gemmGrid(V / 16);                                 // 625 tiles
  const int  vhBlocks   = (V * H + 255) / 256;                 // 5000
  const int  waveBlocks = (EDGES_TOT + 7) / 8;                 // wave-per-edge, 8 waves/block
  const int  expBlocks  = (EDGES_TOT + 255) / 256;

  for (int f = 0; f < N_FRAMES; ++f) {
    const float* xf   = x   + (size_t)f * V * C_IN;
    float*       outf = out + (size_t)f * V * H;

    // ---- Layer 1 ----
    gemm_dual_wmma<C_IN><<<gemmGrid, blk, 0, stream>>>(xf, Wl1, Wr1, xl, xr);
    init_buffers<<<vhBlocks, blk, 0, stream>>>(mmax, denom, acc);
    edge_logit  <<<waveBlocks, blk, 0, stream>>>(xl, xr, ei, att1, evals, mmax);
    edge_exp    <<<expBlocks, blk, 0, stream>>>(evals, mmax, ei, avals, denom);
    edge_scatter<<<waveBlocks, blk, 0, stream>>>(xl, ei, avals, denom, acc);
    bias_relu   <<<vhBlocks, blk, 0, stream>>>(acc, b1, h1);

    // ---- Layer 2 ----
    gemm_dual_wmma<H><<<gemmGrid, blk, 0, stream>>>(h1, Wl2, Wr2, xl, xr);
    init_buffers<<<vhBlocks, blk, 0, stream>>>(mmax, denom, acc);
    edge_logit  <<<waveBlocks, blk, 0, stream>>>(xl, xr, ei, att2, evals, mmax);
    edge_exp    <<<expBlocks, blk, 0, stream>>>(evals, mmax, ei, avals, denom);
    edge_scatter<<<waveBlocks, blk, 0, stream>>>(xl, ei, avals, denom, acc);
    bias_relu   <<<vhBlocks, blk, 0, stream>>>(acc, b2, outf);
  }
}